// FilterNetworkDownsample_6408091206052
// MI455X (gfx1250) — compile-verified
//
#include <hip/hip_runtime.h>
#include <cstdint>

#define BB 8
#define CC 3
#define HH 512
#define WW 512
#define TILE_W 32
#define TILE_H 8
#define HALO_W (TILE_W + 4)   // 36
#define HALO_H (TILE_H + 4)   // 12
#define NHALO (CC * HALO_H * HALO_W)  // 1296 floats = 5184 B LDS

#if __has_builtin(__builtin_amdgcn_global_load_async_to_lds_b32)
#define HAVE_ASYNC_LDS 1
#else
#define HAVE_ASYNC_LDS 0
#endif

typedef __attribute__((address_space(1))) int* gas_int_ptr;
typedef __attribute__((address_space(3))) int* las_int_ptr;

// tanh(x) = (e^{2x}-1)/(e^{2x}+1) = 1 - 2/(exp2(2*log2e*x)+1)
// v_exp_f32 + v_rcp_f32: ~4 VALU ops, correct saturation at +/-1.
__device__ __forceinline__ float fast_tanh(float x) {
#if __has_builtin(__builtin_amdgcn_exp2f) && __has_builtin(__builtin_amdgcn_rcpf)
    float e = __builtin_amdgcn_exp2f(x * 2.8853900817779268f);
    return 1.0f - 2.0f * __builtin_amdgcn_rcpf(e + 1.0f);
#else
    float e = __expf(2.0f * x);
    return 1.0f - 2.0f / (e + 1.0f);
#endif
}

__global__ __launch_bounds__(256) void filter_ds_kernel(
    const float* __restrict__ x, const float* __restrict__ kp,
    float* __restrict__ result, float* __restrict__ down)
{
    __shared__ float tile[NHALO];  // x halo; reused later as result staging (768 floats)

    const int tid = threadIdx.x;        // 0..255
    const int lx  = tid & 31;
    const int ly  = tid >> 5;           // wave index == tile row
    const int gx0 = blockIdx.x * TILE_W;
    const int gy0 = blockIdx.y * TILE_H;
    const int b   = blockIdx.z;

    // Prefetch every kernel-param cache line this tile needs (25 planes x 8 rows,
    // each tile row = one aligned 128B line) so the weight stream overlaps staging.
    if (tid < 25 * TILE_H) {
        int k = tid >> 3, r = tid & 7;
        __builtin_prefetch(&kp[(((size_t)b * 25 + k) * HH + gy0 + r) * WW + gx0], 0, 3);
    }

    // ---- stage zero-padded x halo (3 channels) into LDS ----
#if HAVE_ASYNC_LDS
    for (int idx = tid; idx < NHALO; idx += 256) tile[idx] = 0.0f;
    __syncthreads();
    for (int idx = tid; idx < NHALO; idx += 256) {
        int c   = idx / (HALO_H * HALO_W);
        int rem = idx - c * (HALO_H * HALO_W);
        int r   = rem / HALO_W;
        int col = rem - r * HALO_W;
        int gy  = gy0 - 2 + r;
        int gx  = gx0 - 2 + col;
        if ((unsigned)gy < HH && (unsigned)gx < WW) {
            const float* gp = &x[(((size_t)b * CC + c) * HH + gy) * WW + gx];
            __builtin_amdgcn_global_load_async_to_lds_b32(
                (gas_int_ptr)(void*)gp,
                (las_int_ptr)(void*)&tile[idx],
                0, 0);
        }
    }
#if __has_builtin(__builtin_amdgcn_s_wait_asynccnt)
    __builtin_amdgcn_s_wait_asynccnt(0);
#else
    asm volatile("s_wait_asynccnt 0" ::: "memory");
#endif
    __syncthreads();
#else
    for (int idx = tid; idx < NHALO; idx += 256) {
        int c   = idx / (HALO_H * HALO_W);
        int rem = idx - c * (HALO_H * HALO_W);
        int r   = rem / HALO_W;
        int col = rem - r * HALO_W;
        int gy  = gy0 - 2 + r;
        int gx  = gx0 - 2 + col;
        float v = 0.0f;
        if ((unsigned)gy < HH && (unsigned)gx < WW)
            v = x[(((size_t)b * CC + c) * HH + gy) * WW + gx];
        tile[idx] = v;
    }
    __syncthreads();
#endif

    const int px = gx0 + lx;
    const int py = gy0 + ly;

    // ---- load + tanh the 25 per-pixel weights (coalesced 128B per wave per plane) ----
    float w[25];
    const float* kpp = kp + (((size_t)b * 25) * HH + py) * WW + px;
#pragma unroll
    for (int k = 0; k < 25; ++k)
        w[k] = fast_tanh(kpp[(size_t)k * HH * WW]);

    // ---- 5x5 gather-MAC for all 3 channels (conflict-free LDS reads) ----
    float acc0 = 0.0f, acc1 = 0.0f, acc2 = 0.0f;
#pragma unroll
    for (int i = 0; i < 5; ++i) {
#pragma unroll
        for (int j = 0; j < 5; ++j) {
            float wk = w[i * 5 + j];
            int base = (ly + i) * HALO_W + (lx + j);
            acc0 = fmaf(tile[0 * HALO_H * HALO_W + base], wk, acc0);
            acc1 = fmaf(tile[1 * HALO_H * HALO_W + base], wk, acc1);
            acc2 = fmaf(tile[2 * HALO_H * HALO_W + base], wk, acc2);
        }
    }

    // ---- write result (coalesced) ----
    const size_t plane = (size_t)HH * WW;
    size_t ro = ((size_t)b * CC * HH + py) * WW + px;
    result[ro]             = acc0;
    result[ro + plane]     = acc1;
    result[ro + 2 * plane] = acc2;

    // ---- 2x2 average pool via LDS staging (reuse halo buffer) ----
    __syncthreads();  // all LDS x-reads done before overwrite
    tile[(0 * TILE_H + ly) * TILE_W + lx] = acc0;
    tile[(1 * TILE_H + ly) * TILE_W + lx] = acc1;
    tile[(2 * TILE_H + ly) * TILE_W + lx] = acc2;
    __syncthreads();

    if (tid < CC * (TILE_H / 2) * (TILE_W / 2)) {  // 192 threads
        int c   = tid >> 6;          // /64
        int rem = tid & 63;
        int pyq = rem >> 4;          // 0..3
        int pxq = rem & 15;          // 0..15
        const float* rp = tile + (c * TILE_H + 2 * pyq) * TILE_W + 2 * pxq;
        float s = 0.25f * (rp[0] + rp[1] + rp[TILE_W] + rp[TILE_W + 1]);
        down[((size_t)(b * CC + c) * (HH / 2) + (gy0 >> 1) + pyq) * (WW / 2)
             + (gx0 >> 1) + pxq] = s;
    }
}

extern "C" void kernel_launch(void* const* d_in, const int* in_sizes, int n_in,
                              void* d_out, int out_size, void* d_ws, size_t ws_size,
                              hipStream_t stream) {
    (void)in_sizes; (void)n_in; (void)out_size; (void)d_ws; (void)ws_size;
    const float* x  = (const float*)d_in[0];
    const float* kp = (const float*)d_in[1];
    float* result = (float*)d_out;
    float* down   = (float*)d_out + (size_t)BB * CC * HH * WW;
    dim3 grid(WW / TILE_W, HH / TILE_H, BB);
    filter_ds_kernel<<<grid, 256, 0, stream>>>(x, kp, result, down);
}